// GINmollipo_82815559401960
// MI455X (gfx1250) — compile-verified
//
#include <hip/hip_runtime.h>
#include <hip/hip_bf16.h>

// ---------------------------------------------------------------------------
// GIN (3 layers) + pooling + head for MI455X (gfx1250, wave32, WMMA).
//
// Sizes (from reference): N=50000 nodes, E=800000 edges, C=HID=128, EDIM=16,
// G=512 graphs. Memory-bound: fuse edge linear + gather + relu + scatter-add
// (saves ~820MB/layer), keep agg/pooled L2-resident for atomics, run all
// GEMMs in fp32 on V_WMMA_F32_16X16X4_F32 (compute is free at this roofline).
// ---------------------------------------------------------------------------

#define N_NODES 50000
#define N_EDGES 800000
#define CH      128
#define EDIM    16
#define NGRAPH  512
#define BN_EPS  1e-5f

typedef float v2f __attribute__((ext_vector_type(2)));
typedef float v8f __attribute__((ext_vector_type(8)));

__device__ __forceinline__ v8f wmma_f32(v2f a, v2f b, v8f c) {
  // 16x16 (M,N) x K=4, fp32 in / fp32 accum. 8-arg VOP3P form:
  // (neg_a, A, neg_b, B, c_mod, C, reuse_a, reuse_b)
  return __builtin_amdgcn_wmma_f32_16x16x4_f32(false, a, false, b, (short)0, c,
                                               false, false);
}

// ---------------------------------------------------------------------------
// Utility: zero a float buffer.
// ---------------------------------------------------------------------------
__global__ void zero_kernel(float* __restrict__ p, int n) {
  int i = blockIdx.x * 256 + threadIdx.x;
  if (i < n) p[i] = 0.0f;
}

// ---------------------------------------------------------------------------
// Fused edge kernel:
//   e   = edge_attr @ We + be            (WMMA, K=16)
//   msg = relu(x[src] + e)               (gather epilogue)
//   agg[dst] += msg                      (L2-resident f32 atomics)
// One wave handles 16 edges; block = 4 waves = 64 edges. E % 64 == 0.
// ---------------------------------------------------------------------------
__global__ void gin_edge_kernel(const float* __restrict__ xin,
                                const int*   __restrict__ eidx,
                                const float* __restrict__ eattr,
                                const float* __restrict__ We,
                                const float* __restrict__ be,
                                float*       __restrict__ agg) {
  __shared__ float sWe[EDIM * CH];  // 8 KB, shared by all 4 waves

  const int tid = threadIdx.x;
  // cooperative LDS fill (b128 loads/stores)
  for (int i = tid * 4; i < EDIM * CH; i += 128 * 4) {
    float4 v = *(const float4*)(We + i);
    *(float4*)(sWe + i) = v;
  }
  __syncthreads();

  const int wave = tid >> 5;
  const int lane = tid & 31;
  const int lo   = lane & 15;
  const int hi   = lane >> 4;
  const int e0   = (blockIdx.x * 4 + wave) * 16;

  // speculative prefetch of the next block's edge_attr rows
  if (e0 + 64 < N_EDGES)
    __builtin_prefetch(eattr + (e0 + 64) * EDIM, 0, 0);

  // A-frags: 16x4 f32 layout. lane m(0..15): VGPR{0,1}=K{0,1}; lanes 16..31: K{2,3}.
  const float* arow = eattr + (e0 + lo) * EDIM;
  v2f a[4];
#pragma unroll
  for (int kb = 0; kb < 4; ++kb) {
    a[kb].x = arow[kb * 4 + 2 * hi];
    a[kb].y = arow[kb * 4 + 2 * hi + 1];
  }

  // per-row src/dst (C/D row M = j + 8*hi)
  int srcv[8], dstv[8];
#pragma unroll
  for (int j = 0; j < 8; ++j) {
    int ed  = e0 + j + 8 * hi;
    srcv[j] = eidx[ed];
    dstv[j] = eidx[N_EDGES + ed];
  }

#pragma unroll
  for (int t = 0; t < 8; ++t) {        // 8 tiles of 16 output channels
    const int ch = t * 16 + lo;
    const float bv = be[ch];
    v8f acc;
#pragma unroll
    for (int j = 0; j < 8; ++j) acc[j] = bv;   // bias broadcast along M

#pragma unroll
    for (int kb = 0; kb < 4; ++kb) {   // K = 16 as 4 chained WMMAs
      v2f b;
      b.x = sWe[(kb * 4 + 2 * hi) * CH + ch];
      b.y = sWe[(kb * 4 + 2 * hi + 1) * CH + ch];
      acc = wmma_f32(a[kb], b, acc);
    }

#pragma unroll
    for (int j = 0; j < 8; ++j) {      // epilogue: gather + relu + scatter
      float msg = acc[j] + xin[(long)srcv[j] * CH + ch];
      msg = msg > 0.0f ? msg : 0.0f;
      atomicAdd(agg + (long)dstv[j] * CH + ch, msg);
    }
  }
}

// ---------------------------------------------------------------------------
// Fused node MLP kernel (per layer):
//   out = agg + (1+eps)*x
//   h   = relu(bn_a(out @ W1 + b1))
//   h   = bn_b(relu(h @ W2 + b2))
//   y   = bn_outer(h)                      -> xout
//   pooled[batch[n]] += y                  (fused global_add_pool)
// One wave per 16 nodes; block = 8 waves = 128 nodes.
// LDS: 64KB weight stage (W1 then W2) + 64KB activation transpose.
// ---------------------------------------------------------------------------
struct GinLayerParams {
  const float *We, *be, *eps;
  const float *W1, *b1;
  const float *bnaG, *bnaB, *bnaM, *bnaV;
  const float *W2, *b2;
  const float *bnbG, *bnbB, *bnbM, *bnbV;
  const float *bnoG, *bnoB, *bnoM, *bnoV;
};

__global__ void gin_mlp_kernel(const float* __restrict__ xin,
                               const float* __restrict__ agg,
                               GinLayerParams p,
                               float* __restrict__ xout,
                               float* __restrict__ pooledL,
                               const int* __restrict__ batch) {
  extern __shared__ float smem[];
  float* sW = smem;              // 128x128 weights (staged W1 then W2)
  float* sH = smem + CH * CH;    // 8 waves x 16 rows x 128 cols

  const int tid  = threadIdx.x;
  const int wave = tid >> 5;
  const int lane = tid & 31;
  const int lo   = lane & 15;
  const int hi   = lane >> 4;
  const int n0   = (blockIdx.x * 8 + wave) * 16;

  // ---- stage W1 ----
  for (int i = tid * 4; i < CH * CH; i += 256 * 4) {
    float4 v = *(const float4*)(p.W1 + i);
    *(float4*)(sW + i) = v;
  }
  __syncthreads();

  const float eps1 = 1.0f + p.eps[0];

  // A-frags for GEMM1: rows = nodes (lane lo), cols 4*kk + 2*hi {+1}
  int row = n0 + lo;
  if (row >= N_NODES) row = N_NODES - 1;   // clamp; stores are guarded below
  const float* aggr = agg + (long)row * CH;
  const float* xinr = xin + (long)row * CH;

  v2f a[32];
#pragma unroll
  for (int kk = 0; kk < 32; ++kk) {
    int c   = kk * 4 + 2 * hi;
    a[kk].x = aggr[c]     + eps1 * xinr[c];
    a[kk].y = aggr[c + 1] + eps1 * xinr[c + 1];
  }

  // ---- GEMM1 (K=128) + bn_a + relu -> LDS transpose buffer ----
  float* hrow = sH + wave * 16 * CH;
#pragma unroll
  for (int t = 0; t < 8; ++t) {
    const int ch = t * 16 + lo;
    const float bv = p.b1[ch];
    v8f acc;
#pragma unroll
    for (int j = 0; j < 8; ++j) acc[j] = bv;
#pragma unroll
    for (int kk = 0; kk < 32; ++kk) {
      v2f b;
      int r = kk * 4 + 2 * hi;
      b.x = sW[r * CH + ch];
      b.y = sW[(r + 1) * CH + ch];
      acc = wmma_f32(a[kk], b, acc);
    }
    const float g  = p.bnaG[ch], bb = p.bnaB[ch];
    const float mm = p.bnaM[ch];
    const float iv = rsqrtf(p.bnaV[ch] + BN_EPS);
#pragma unroll
    for (int j = 0; j < 8; ++j) {
      float v = g * (acc[j] - mm) * iv + bb;   // bn_a
      v = v > 0.0f ? v : 0.0f;                 // relu
      hrow[(j + 8 * hi) * CH + ch] = v;        // C-layout -> row-major in LDS
    }
  }
  __syncthreads();           // all waves done reading W1 and writing sH

  // ---- stage W2 over W1 ----
  for (int i = tid * 4; i < CH * CH; i += 256 * 4) {
    float4 v = *(const float4*)(p.W2 + i);
    *(float4*)(sW + i) = v;
  }
  __syncthreads();

  // A-frags for GEMM2 from LDS (transposed read)
#pragma unroll
  for (int kk = 0; kk < 32; ++kk) {
    int c   = kk * 4 + 2 * hi;
    a[kk].x = hrow[lo * CH + c];
    a[kk].y = hrow[lo * CH + c + 1];
  }

  // ---- GEMM2 (K=128) + relu + bn_b + bn_outer -> store + fused pooling ----
#pragma unroll
  for (int t = 0; t < 8; ++t) {
    const int ch = t * 16 + lo;
    const float bv = p.b2[ch];
    v8f acc;
#pragma unroll
    for (int j = 0; j < 8; ++j) acc[j] = bv;
#pragma unroll
    for (int kk = 0; kk < 32; ++kk) {
      v2f b;
      int r = kk * 4 + 2 * hi;
      b.x = sW[r * CH + ch];
      b.y = sW[(r + 1) * CH + ch];
      acc = wmma_f32(a[kk], b, acc);
    }
    const float gb = p.bnbG[ch], bbb = p.bnbB[ch];
    const float mb = p.bnbM[ch];
    const float ib = rsqrtf(p.bnbV[ch] + BN_EPS);
    const float go = p.bnoG[ch], bo = p.bnoB[ch];
    const float mo = p.bnoM[ch];
    const float io = rsqrtf(p.bnoV[ch] + BN_EPS);
#pragma unroll
    for (int j = 0; j < 8; ++j) {
      float v = acc[j];
      v = v > 0.0f ? v : 0.0f;                 // relu
      v = gb * (v - mb) * ib + bbb;            // bn_b
      v = go * (v - mo) * io + bo;             // outer bn (bn1/bn2/bn3)
      const int r = n0 + j + 8 * hi;
      if (r < N_NODES) {
        xout[(long)r * CH + ch] = v;
        const int g = batch[r];
        atomicAdd(pooledL + (long)g * CH + ch, v);   // fused global_add_pool
      }
    }
  }
}

// ---------------------------------------------------------------------------
// Head: h = leaky_relu(bn4(concat(p1,p2,p3) @ W_l1 + b_l1));  out = h @ W_l2 + b_l2
// Tiny (512x384) -> VALU kernels; compute is negligible at this roofline.
// ---------------------------------------------------------------------------
__global__ void head1_kernel(const float* __restrict__ pooled,  // [3][G][128]
                             const float* __restrict__ Wl1,     // [384][384]
                             const float* __restrict__ bl1,
                             const float* __restrict__ g4, const float* __restrict__ b4,
                             const float* __restrict__ m4, const float* __restrict__ v4,
                             float* __restrict__ hbuf) {        // [G][384]
  const int idx = blockIdx.x * 256 + threadIdx.x;
  if (idx >= NGRAPH * 3 * CH) return;
  const int g = idx / (3 * CH);
  const int c = idx - g * (3 * CH);

  float acc = bl1[c];
#pragma unroll
  for (int part = 0; part < 3; ++part) {
    const float* hp = pooled + (part * NGRAPH + g) * CH;
    const float* wp = Wl1 + part * CH * (3 * CH) + c;
#pragma unroll 8
    for (int kk = 0; kk < CH; ++kk) acc += hp[kk] * wp[kk * (3 * CH)];
  }
  float v = g4[c] * (acc - m4[c]) * rsqrtf(v4[c] + BN_EPS) + b4[c];  // bn4
  v = v >= 0.0f ? v : 0.01f * v;                                     // leaky_relu
  hbuf[idx] = v;
}

__global__ void head2_kernel(const float* __restrict__ hbuf,
                             const float* __restrict__ Wl2,   // [384]
                             const float* __restrict__ bl2,   // [1]
                             float* __restrict__ out) {       // [G]
  const int g = blockIdx.x * 256 + threadIdx.x;
  if (g >= NGRAPH) return;
  float acc = bl2[0];
  const float* h = hbuf + g * (3 * CH);
#pragma unroll 8
  for (int k = 0; k < 3 * CH; ++k) acc += h[k] * Wl2[k];
  out[g] = acc;
}

// ---------------------------------------------------------------------------
// Launch. Input flattening (setup_inputs insertion order, params depth-first):
//  0:x 1:edge_index 2:edge_attr 3:batch, then per gin layer (stride 19):
//  We,be,eps,W1,b1,bn_a(g,b,m,v),W2,b2,bn_b(g,b,m,v),bn_outer(g,b,m,v);
//  61:W_l1 62:b_l1 63..66:bn4 67:W_l2 68:b_l2
// ---------------------------------------------------------------------------
extern "C" void kernel_launch(void* const* d_in, const int* in_sizes, int n_in,
                              void* d_out, int out_size, void* d_ws, size_t ws_size,
                              hipStream_t stream) {
  const float* x     = (const float*)d_in[0];
  const int*   eidx  = (const int*)d_in[1];
  const float* eattr = (const float*)d_in[2];
  const int*   batch = (const int*)d_in[3];
  auto P = [&](int i) { return (const float*)d_in[i]; };

  const int NF = N_NODES * CH;          // 6,400,000 floats
  float* ws     = (float*)d_ws;
  float* agg    = ws;
  float* xb[3]  = {ws + NF, ws + 2 * NF, ws + 3 * NF};
  float* pooled = ws + 4 * NF;                       // [3][G][128]
  float* hbuf   = pooled + 3 * NGRAPH * CH;          // [G][384]

  zero_kernel<<<(3 * NGRAPH * CH + 255) / 256, 256, 0, stream>>>(pooled,
                                                                 3 * NGRAPH * CH);

  const float* xcur = x;
  for (int L = 0; L < 3; ++L) {
    const int base = 4 + L * 19;
    GinLayerParams p = {
        P(base + 0),  P(base + 1),  P(base + 2),              // We, be, eps
        P(base + 3),  P(base + 4),                            // W1, b1
        P(base + 5),  P(base + 6),  P(base + 7),  P(base + 8),  // bn_a
        P(base + 9),  P(base + 10),                           // W2, b2
        P(base + 11), P(base + 12), P(base + 13), P(base + 14), // bn_b
        P(base + 15), P(base + 16), P(base + 17), P(base + 18)  // bn1/2/3
    };

    zero_kernel<<<(NF + 255) / 256, 256, 0, stream>>>(agg, NF);
    gin_edge_kernel<<<N_EDGES / 64, 128, 0, stream>>>(xcur, eidx, eattr,
                                                      p.We, p.be, agg);
    const int mlp_blocks = ((N_NODES + 15) / 16 + 7) / 8;   // 391
    gin_mlp_kernel<<<mlp_blocks, 256, 2 * CH * CH * sizeof(float), stream>>>(
        xcur, agg, p, xb[L], pooled + L * NGRAPH * CH, batch);
    xcur = xb[L];
  }

  head1_kernel<<<(NGRAPH * 3 * CH + 255) / 256, 256, 0, stream>>>(
      pooled, P(61), P(62), P(63), P(64), P(65), P(66), hbuf);
  head2_kernel<<<(NGRAPH + 255) / 256, 256, 0, stream>>>(hbuf, P(67), P(68),
                                                         (float*)d_out);
}